// WarpingOutput_2130303779368
// MI455X (gfx1250) — compile-verified
//
#include <hip/hip_runtime.h>

// Dense image warp (bilinear, edge-clamped), fused for both (frame, flow) pairs.
// out0 = warp(frame_0, f_10) ; out1 = warp(frame_1, f_01), concatenated in d_out.
//
// Memory-bound: ~268 MB of traffic -> ~11.5 us floor at 23.3 TB/s.
// Strategy: 2 pixels/thread so the streamed flow is one b128 NT load and the
// streamed output is 24 B/thread of NT stores; frame gathers use default
// (cached) policy so both 50 MB frames stay resident in the 192 MB L2.

typedef float v4f __attribute__((ext_vector_type(4)));

#define IMG_H 512
#define IMG_W 512

__global__ __launch_bounds__(256) void warp_bilinear2_kernel(
    const float* __restrict__ frame0,
    const float* __restrict__ frame1,
    const float* __restrict__ f01,
    const float* __restrict__ f10,
    float* __restrict__ out,
    int N)  // N = B*H*W pixels per image (even, multiple of 512)
{
    int q = (int)blockIdx.x * 256 + (int)threadIdx.x;  // pixel-pair index
    if (q >= N) return;                                // N pairs total (2N pixels / 2)

    const int base = q << 1;           // first pixel of the pair, in [0, 2N)
    const bool second = base >= N;     // block-uniform: N is a multiple of 512
    const int p0 = second ? (base - N) : base;

    const float* __restrict__ img  = second ? frame1 : frame0;
    const float* __restrict__ flow = second ? f01 : f10;
    float* __restrict__ o =
        out + (second ? (size_t)3 * (size_t)N : (size_t)0) + (size_t)3 * (size_t)p0;

    // Two flow vectors (dy0,dx0,dy1,dx1) in one non-temporal 128-bit load.
    v4f f = __builtin_nontemporal_load((const v4f*)flow + (p0 >> 1));

    float res[6];
#pragma unroll
    for (int j = 0; j < 2; ++j) {
        const int p = p0 + j;
        const int x = p & (IMG_W - 1);
        const int y = (p >> 9) & (IMG_H - 1);
        const int b = p >> 18;

        const float fy = (j == 0) ? f.x : f.z;
        const float fx = (j == 0) ? f.y : f.w;
        const float qy = (float)y - fy;
        const float qx = (float)x - fx;

        // TFA interpolate_bilinear: floor clipped to [0, size-2], alpha to [0,1].
        float y0f = fminf(fmaxf(floorf(qy), 0.0f), (float)(IMG_H - 2));
        float x0f = fminf(fmaxf(floorf(qx), 0.0f), (float)(IMG_W - 2));
        float ay = fminf(fmaxf(qy - y0f, 0.0f), 1.0f);
        float ax = fminf(fmaxf(qx - x0f, 0.0f), 1.0f);
        int y0 = (int)y0f;
        int x0 = (int)x0f;

        // Corner pairs: 6 contiguous floats per row (24 B), merged into wide loads.
        const float* __restrict__ rt =
            img + (size_t)((((b << 9) + y0) << 9) + x0) * 3;  // tl|tr
        const float* __restrict__ rb = rt + IMG_W * 3;        // bl|br

#pragma unroll
        for (int c = 0; c < 3; ++c) {
            float tl = rt[c];
            float tr = rt[3 + c];
            float bl = rb[c];
            float br = rb[3 + c];
            float top = fmaf(ax, tr - tl, tl);
            float bot = fmaf(ax, br - bl, bl);
            res[3 * j + c] = fmaf(ay, bot - top, top);
        }
    }

    // 24 B of contiguous NT stores per thread (mergeable to b128 + b64).
#pragma unroll
    for (int c = 0; c < 6; ++c) {
        __builtin_nontemporal_store(res[c], o + c);
    }
}

extern "C" void kernel_launch(void* const* d_in, const int* in_sizes, int n_in,
                              void* d_out, int out_size, void* d_ws, size_t ws_size,
                              hipStream_t stream) {
    // setup_inputs order: frame_0, frame_1, f_01, f_10
    const float* frame0 = (const float*)d_in[0];
    const float* frame1 = (const float*)d_in[1];
    const float* f01    = (const float*)d_in[2];
    const float* f10    = (const float*)d_in[3];
    float* out = (float*)d_out;

    const int N = in_sizes[3] / 2;  // B*H*W pixels per image
    if (N <= 0) return;
    const int pairs = N;            // 2N pixels, 2 per thread
    const int blocks = (pairs + 255) / 256;

    hipLaunchKernelGGL(warp_bilinear2_kernel, dim3(blocks), dim3(256), 0, stream,
                       frame0, frame1, f01, f10, out, N);
}